// SpikingExecutiveKernel_40192303956594
// MI455X (gfx1250) — compile-verified
//
#include <hip/hip_runtime.h>
#include <stdint.h>

// Problem constants (from reference setup): B=4, L=512, D=2048, N=16
#define BB     4
#define LL     512
#define DD     2048
#define NN     16
#define CT     128      // timesteps staged per LDS chunk
#define NC     (LL/CT)  // 4 chunks
#define DBLK   16       // d-values per block
#define NTHR   256      // threads per block (16 d * 16 n) = 8 wave32

// Truncate a generic (flat) pointer to the 32-bit LDS byte offset.
// Per CDNA5 ISA 10.2: LDS-aperture flat addresses use their low 32 bits as the LDS address.
__device__ __forceinline__ uint32_t lds_off_u32(const void* p) {
  return (uint32_t)(uintptr_t)p;
}

__device__ __forceinline__ void async_ld_b128(uint32_t lds_byte_off, const void* gptr) {
  uint64_t ga = (uint64_t)(uintptr_t)gptr;
  // GV mode: VDST = LDS byte address, VADDR = 64-bit global address, no SADDR.
  asm volatile("global_load_async_to_lds_b128 %0, %1, off"
               :: "v"(lds_byte_off), "v"(ga)
               : "memory");
}

__device__ __forceinline__ void wait_async0() {
#if __has_builtin(__builtin_amdgcn_s_wait_asynccnt)
  __builtin_amdgcn_s_wait_asynccnt(0);
#else
  asm volatile("s_wait_asynccnt 0" ::: "memory");
#endif
}

__global__ __launch_bounds__(NTHR)
void spiking_scan_kernel(const float* __restrict__ s,
                         const int*   __restrict__ Delta,
                         const float* __restrict__ Bt,
                         const float* __restrict__ Ct,
                         const float* __restrict__ Alog,
                         const float* __restrict__ h0,
                         float* __restrict__ out)
{
  // Double-buffered staging: 4 arrays x 2 buffers x 8KB = 64 KB LDS total.
  __shared__ __align__(16) float lB[2][CT * NN];
  __shared__ __align__(16) float lC[2][CT * NN];
  __shared__ __align__(16) float lS[2][CT * DBLK];
  __shared__ __align__(16) int   lG[2][CT * DBLK];

  const int tid  = threadIdx.x;
  const int blk  = blockIdx.x;
  const int b    = blk >> 7;          // 128 d-blocks per batch
  const int dblk = blk & 127;
  const int d0   = dblk * DBLK;
  const int dl   = tid >> 4;          // 0..15  (d within block)
  const int n    = tid & 15;          // 0..15
  const int d    = d0 + dl;

  // Per-(d,n) constant: two_K = exp2(round(exp(A_log))). rintf == round-half-even (jnp.round).
  const float twoK = exp2f(rintf(expf(Alog[d * NN + n])));

  // Initial state (harness passes zeros, but honor h0 for fidelity).
  float h = h0[((size_t)b * DD + d) * NN + n];

  float* ybase = out + (size_t)b * LL * DD * NN + (size_t)d * NN + n;
  const size_t ystride = (size_t)DD * NN;   // per timestep

  // Stage chunk c into buffer (c&1). Issues 8 async b128 per thread; no wait here.
  auto stage = [&](int c) {
    const int p = c & 1;
    const uint32_t lB_off = lds_off_u32(&lB[p][0]);
    const uint32_t lC_off = lds_off_u32(&lC[p][0]);
    const uint32_t lS_off = lds_off_u32(&lS[p][0]);
    const uint32_t lG_off = lds_off_u32(&lG[p][0]);

    // B_t / C_t chunk: contiguous CT*NN floats (8 KB each) -> 2 rounds of 256x16B.
    const char* gb = (const char*)(Bt + ((size_t)b * LL + (size_t)c * CT) * NN);
    const char* gc = (const char*)(Ct + ((size_t)b * LL + (size_t)c * CT) * NN);
#pragma unroll
    for (int r = 0; r < (CT * NN * 4) / (NTHR * 16); ++r) {
      uint32_t off = (uint32_t)(r * NTHR + tid) * 16u;
      async_ld_b128(lB_off + off, gb + off);
      async_ld_b128(lC_off + off, gc + off);
    }

    // s / Delta chunk: CT rows of 64B (16 d-values), row stride D*4 bytes -> 2 rounds.
#pragma unroll
    for (int r = 0; r < (CT * 4) / NTHR; ++r) {
      int idx = r * NTHR + tid;        // 0..511
      int tl  = idx >> 2;              // t within chunk
      int q   = idx & 3;               // which 16B quad of the 64B row
      size_t gelem = (((size_t)b * LL + (size_t)c * CT + tl) * DD + d0 + q * 4);
      uint32_t loff = (uint32_t)((tl * DBLK + q * 4) * 4);
      async_ld_b128(lS_off + loff, s     + gelem);
      async_ld_b128(lG_off + loff, Delta + gelem);
    }
  };

  // Software pipeline: stage(c+1) overlaps with compute(c).
  stage(0);

  for (int c = 0; c < NC; ++c) {
    const int p = c & 1;

    wait_async0();     // my staging of chunk c is complete (issued last iteration or above)
    __syncthreads();   // all waves: chunk c staged AND compute(c-1) done (protects both buffers)

    if (c + 1 < NC) stage(c + 1);   // async fill of the other buffer; overlaps compute below

    // ---- Serial recurrence over this chunk, fed entirely from LDS ----
    float* ychunk = ybase + (size_t)(c * CT) * ystride;
#pragma unroll 8
    for (int tl = 0; tl < CT; ++tl) {
      float sv = lS[p][tl * DBLK + dl];
      int   gv = lG[p][tl * DBLK + dl];
      float bv = lB[p][tl * NN + n];
      float cv = lC[p][tl * NN + n];

      bool  gate = (gv != 0);
      float Abar = gate ? twoK : 1.0f;          // where(gate==1, two_K, 1)
      float bs   = gate ? bv * sv : 0.0f;       // (gate * b_t) * s_t, rounded once
      float pre  = __builtin_fmaf(Abar, h, bs); // exact: Abar*h exact (pow2 * {0,1})
      float hn   = (pre > 1.0f) ? 1.0f : 0.0f;  // _spike
      float y    = (cv * hn > 1.0f) ? 1.0f : 0.0f;

      ychunk[(size_t)tl * ystride] = y;
      h = hn;
    }
  }

  // h_final appended after y in d_out
  const size_t YTOT = (size_t)BB * LL * DD * NN;
  out[YTOT + ((size_t)b * DD + d) * NN + n] = h;
}

extern "C" void kernel_launch(void* const* d_in, const int* in_sizes, int n_in,
                              void* d_out, int out_size, void* d_ws, size_t ws_size,
                              hipStream_t stream) {
  const float* s     = (const float*)d_in[0];
  const int*   Delta = (const int*)d_in[1];
  const float* Bt    = (const float*)d_in[2];
  const float* Ct    = (const float*)d_in[3];
  const float* Alog  = (const float*)d_in[4];
  const float* h0    = (const float*)d_in[5];
  float* out = (float*)d_out;

  dim3 grid(BB * (DD / DBLK));   // 4 * 128 = 512 blocks
  dim3 block(NTHR);              // 256 threads = 8 wave32
  spiking_scan_kernel<<<grid, block, 0, stream>>>(s, Delta, Bt, Ct, Alog, h0, out);
}